// CTCLayer_3186865734014
// MI455X (gfx1250) — compile-verified
//
#include <hip/hip_runtime.h>
#include <hip/hip_bf16.h>

// Problem constants from the reference: B,T,C,L = 512,512,128,64
#define B_    512
#define T_    512
#define C_    128
#define L_    64
#define S_    129          // 2*L+1 extended states, blank = C-1
#define SPL   5            // states per lane (32*5 = 160 >= 129)
#define DEPTH 8            // async rows in flight (ASYNCcnt pipeline depth)
#define RING  16           // LDS ring rows (> DEPTH: no read/write slot hazard)
#define PFA   32           // L2 prefetch distance (rows)
#define BLANK (C_ - 1)
#define EPSF  1e-7f

#define ROWB  (C_ * 4)     // bytes per probability row (512 B)

__device__ __forceinline__ void async_row_load(unsigned lds_addr, const void* gaddr) {
    // GLOBAL_LOAD_ASYNC_TO_LDS_B128: per lane, 16 B memory -> LDS, tracked by ASYNCcnt
    asm volatile("global_load_async_to_lds_b128 %0, %1, off"
                 :: "v"(lds_addr), "v"((unsigned long long)(uintptr_t)gaddr)
                 : "memory");
}

__global__ __launch_bounds__(32)
void ctc_forward_kernel(const int* __restrict__ y_true,
                        const float* __restrict__ y_pred,
                        float* __restrict__ out) {
    const int b    = blockIdx.x;
    const int lane = threadIdx.x;            // wave32: one wave per block

    __shared__ float ring[RING * C_];        // 16-row ring, 8 KB

    // ---- Per-lane static state info: channel index, allow mask, validity ----
    const int* lab = y_true + (size_t)b * L_;
    int   ch[SPL];
    float am[SPL];                           // allow-skip mask as float (enables FMA)
    float vmask[SPL];
#pragma unroll
    for (int j = 0; j < SPL; ++j) {
        const int s = SPL * lane + j;        // extended state index
        const bool valid = (s < S_);
        vmask[j] = valid ? 1.0f : 0.0f;
        if (valid && (s & 1)) {              // odd state -> label
            const int k = (s - 1) >> 1;
            const int c = lab[k];
            ch[j] = c;
            am[j] = ((k >= 1) && (c != lab[k - 1])) ? 1.0f : 0.0f;
        } else {                             // even/padding state -> blank, no skip
            ch[j] = BLANK;
            am[j] = 0.0f;
        }
    }

    // ---- Async pipeline bases ----
    const char*    gb      = (const char*)(y_pred + (size_t)b * (size_t)(T_ * C_));
    const char*    glane   = gb + lane * 16;                         // this lane's 16B chunk
    const unsigned ldslane = (unsigned)(uintptr_t)ring + (unsigned)(lane * 16);

    // prime DEPTH async row loads (rows 0..DEPTH-1 -> slots 0..DEPTH-1)
#pragma unroll
    for (int d = 0; d < DEPTH; ++d)
        async_row_load(ldslane + (unsigned)(d * ROWB), glane + (size_t)d * ROWB);

    // ---- alpha in registers; virtual alpha_{-1}: only state 0 = 1 ----
    float a[SPL];
#pragma unroll
    for (int j = 0; j < SPL; ++j) a[j] = 0.0f;
    if (lane == 0) a[0] = 1.0f;
    float lz = 0.0f;                         // accumulated log normalizers

#pragma unroll 4
    for (int t = 0; t < T_; ++t) {
        // async loads complete in order: <= DEPTH-1 outstanding => row t resident
        asm volatile("s_wait_asynccnt %0" :: "n"(DEPTH - 1) : "memory");

        // refill: row t+DEPTH into slot (t+DEPTH)%RING (clamped dummy near the end
        // keeps the wait threshold a constant immediate for every step)
        {
            const int tp = (t + DEPTH < T_) ? (t + DEPTH) : (T_ - 1);
            async_row_load(ldslane + (unsigned)(((t + DEPTH) & (RING - 1)) * ROWB),
                           glane + (size_t)tp * ROWB);
        }
        if (t + PFA < T_)                    // keep async engine L2-hot
            __builtin_prefetch(glane + (size_t)(t + PFA) * ROWB, 0, 0);

        // gather per-state probabilities from resident row (ds_load; blank broadcasts)
        const float* rowf = ring + ((t & (RING - 1)) * C_);
        float pe[SPL];
#pragma unroll
        for (int j = 0; j < SPL; ++j)
            pe[j] = (rowf[ch[j]] + EPSF) * vmask[j];

        // neighbor states from lane-1 (last two slots)
        float up1 = __shfl_up(a[SPL - 1], 1, 32);    // s-1 for slot 0
        float up2 = __shfl_up(a[SPL - 2], 1, 32);    // s-2 for slot 0
        if (lane == 0) { up1 = 0.0f; up2 = 0.0f; }

        const float na0 = (a[0] + up1  + am[0] * up2 ) * pe[0];
        const float na1 = (a[1] + a[0] + am[1] * up1 ) * pe[1];
        const float na2 = (a[2] + a[1] + am[2] * a[0]) * pe[2];
        const float na3 = (a[3] + a[2] + am[3] * a[1]) * pe[3];
        const float na4 = (a[4] + a[3] + am[4] * a[2]) * pe[4];
        a[0] = na0; a[1] = na1; a[2] = na2; a[3] = na3; a[4] = na4;

        // rescale every 4 steps: keep linear-domain alphas in range
        if ((t & 3) == 3) {
            float ssum = a[0] + a[1] + a[2] + a[3] + a[4];
#pragma unroll
            for (int off = 16; off >= 1; off >>= 1)
                ssum += __shfl_xor(ssum, off, 32);
            const float inv = 1.0f / ssum;
            a[0] *= inv; a[1] *= inv; a[2] *= inv; a[3] *= inv; a[4] *= inv;
            lz += __logf(ssum);
        }
    }

    // loss = -( log(alpha_T[S-1] + alpha_T[S-2]) + sum log norms )
    // states 127,128 live in lane 25, slots 2,3
    if (lane == 25) {
        const float tail = a[2] + a[3];
        out[b] = -(lz + __logf(tail));
    }
}

extern "C" void kernel_launch(void* const* d_in, const int* in_sizes, int n_in,
                              void* d_out, int out_size, void* d_ws, size_t ws_size,
                              hipStream_t stream) {
    (void)in_sizes; (void)n_in; (void)d_ws; (void)ws_size; (void)out_size;
    const int*   y_true = (const int*)d_in[0];    // [B, L] int32
    const float* y_pred = (const float*)d_in[1];  // [B, T, C] float32 probabilities
    float*       out    = (float*)d_out;          // [B, 1] float32

    ctc_forward_kernel<<<dim3(B_), dim3(32), 0, stream>>>(y_true, y_pred, out);
}